// EdgeEncoding_18691697672326
// MI455X (gfx1250) — compile-verified
//
#include <hip/hip_runtime.h>

typedef __attribute__((ext_vector_type(2))) float v2f;
typedef __attribute__((ext_vector_type(8))) float v8f;

#define PLEN       5      // MAX_PATH_LENGTH
#define EMBED      64
#define NUM_EDGES  512
#define MAX_PATHS  16384
#define MAX_NODES  128
#define BATCH      16
#define WAVES_PB   8      // waves (16-path tiles) per block

__global__ void ee_zero_fill(float* __restrict__ out, int n) {
    int i = blockIdx.x * blockDim.x + threadIdx.x;
    int stride = gridDim.x * blockDim.x;
    for (; i < n; i += stride) out[i] = 0.0f;
}

__global__ __launch_bounds__(256, 2)
void ee_wmma_kernel(const float* __restrict__ edge_features,   // (B,512,64)
                    const float* __restrict__ edge_weights,    // (5,64)
                    const int*   __restrict__ path_cache,      // (B,16384,5)
                    const int*   __restrict__ path_lengths,    // (B,16384)
                    const int*   __restrict__ pairs,           // (B,16384,2)
                    float*       __restrict__ out) {           // (B,128,128)
    __shared__ __align__(16) float lds_w[PLEN * EMBED];        // raw weight table
    __shared__ __align__(16) float lds_d[WAVES_PB * 256];      // per-wave 16x16 D staging

    const int tid  = threadIdx.x;
    const int lane = tid & 31;
    const int wave = tid >> 5;
    const int m    = lane & 15;   // A row / B,D column within tile
    const int hi   = lane >> 4;   // K-half selector

    // Stage the 320-float weight table into LDS once per block.
    for (int i = tid; i < PLEN * EMBED; i += 256) lds_w[i] = edge_weights[i];
    __syncthreads();

    // tile -> (batch, 16-path slice); 1024 tiles per batch
    const int tile = blockIdx.x * WAVES_PB + wave;
    const int b    = tile >> 10;
    const int p    = ((tile & 1023) << 4) + m;

    const long prow = (long)b * MAX_PATHS + p;
    int eidx[PLEN];
#pragma unroll
    for (int k = 0; k < PLEN; ++k) eidx[k] = path_cache[prow * PLEN + k];
    const int plen = path_lengths[prow];
    const int n1   = pairs[prow * 2 + 0];
    const int n2   = pairs[prow * 2 + 1];

    const float* __restrict__ efb = edge_features + (long)b * (NUM_EDGES * EMBED);

    // Accumulate D[path, hypothetical_length] = A(16x320 gathered) x B(320x16 masked weights)
    v8f acc = {0.f, 0.f, 0.f, 0.f, 0.f, 0.f, 0.f, 0.f};
#pragma unroll
    for (int k = 0; k < PLEN; ++k) {
        const float* __restrict__ arow = efb + (eidx[k] << 6);   // this lane's gathered row
        const float* __restrict__ wrow = lds_w + k * EMBED;
        const float  mask = (m >= (k + 1)) ? 1.0f : 0.0f;        // column n=m uses pos k iff k+1<=n
#pragma unroll
        for (int t = 0; t < 16; ++t) {
            const int e = 4 * t + 2 * hi;                        // K-slice element offset
            v2f a  = *(const v2f*)(arow + e);                    // global_load_b64 gather
            v2f wv = *(const v2f*)(wrow + e);                    // ds_load_b64
            wv *= mask;
            acc = __builtin_amdgcn_wmma_f32_16x16x4_f32(
                /*neg_a=*/false, a, /*neg_b=*/false, wv,
                /*c_mod=*/(short)0, acc, /*reuse_a=*/false, /*reuse_b=*/false);
        }
    }

    // Stage D to LDS in (row M, col N) order: lane holds rows v + 8*hi, col m.
    float* dst = lds_d + wave * 256;
#pragma unroll
    for (int v = 0; v < 8; ++v) dst[(v + 8 * hi) * 16 + m] = acc[v];
    __syncthreads();

    // Lanes 0..15: pick column = this path's length, normalize, scatter.
    if (hi == 0) {
        int len = plen;
        if (len < 0) len = 0;
        if (len > PLEN) len = PLEN;
        const float dot   = dst[m * 16 + len];                   // len==0 column is all-zero
        const float denom = (float)(len > 0 ? len : 1);
        out[((long)b * MAX_NODES + n1) * MAX_NODES + n2] = dot / denom;
    }
}

extern "C" void kernel_launch(void* const* d_in, const int* in_sizes, int n_in,
                              void* d_out, int out_size, void* d_ws, size_t ws_size,
                              hipStream_t stream) {
    const float* edge_features = (const float*)d_in[0];
    const float* edge_weights  = (const float*)d_in[1];
    const int*   path_cache    = (const int*)d_in[2];
    const int*   path_lengths  = (const int*)d_in[3];
    const int*   pairs         = (const int*)d_in[4];
    float* out = (float*)d_out;

    // Zero-initialize the scatter target (harness poisons d_out).
    const int n_out = BATCH * MAX_NODES * MAX_NODES;  // 262144
    ee_zero_fill<<<256, 256, 0, stream>>>(out, n_out);

    // 16384 tiles of 16 paths, 8 tiles (waves) per 256-thread block.
    const int n_tiles = (BATCH * MAX_PATHS) / 16;     // 16384
    const int blocks  = n_tiles / WAVES_PB;           // 2048
    ee_wmma_kernel<<<blocks, 256, 0, stream>>>(edge_features, edge_weights,
                                               path_cache, path_lengths, pairs, out);
}